// GCNNet01_60687887893291
// MI455X (gfx1250) — compile-verified
//
#include <hip/hip_runtime.h>

// GCN (1 -> 4 -> 1 -> sigmoid) on MI455X, collapsed to scalar per-edge passes.
// Roofline: ~0.83 GB edge streaming over 3 passes, ~0.2 GFLOP -> HBM-bound,
// ~36 us at 23.3 TB/s. No dense tiles exist after the algebraic collapse, so
// WMMA is inapplicable; the chip-specific wins are memory-path ones:
//   - b128 NT streaming of int64 indices (TH_LOAD_NT keeps L2 for gathers)
//   - global_load_async_to_lds_b128 staging of edge weights (ASYNCcnt)
//   - global_prefetch_b8 on the streams
//   - global_atomic_add_f32 scatter into 4 MB L2-resident accumulators

#define BLK 256

typedef long long v2i64 __attribute__((ext_vector_type(2)));
typedef float     v4f   __attribute__((ext_vector_type(4)));

static __device__ __forceinline__ v2i64 ldnt2(const long long* p) {
    return __builtin_nontemporal_load(reinterpret_cast<const v2i64*>(p));
}
static __device__ __forceinline__ v4f ldnt4f(const float* p) {
    return __builtin_nontemporal_load(reinterpret_cast<const v4f*>(p));
}

// ---------------- node kernels (8-40 MB total traffic, negligible) ----------

__global__ void k_init_deg(float* __restrict__ deg, int n) {
    int i = blockIdx.x * blockDim.x + threadIdx.x;
    if (i < n) deg[i] = 1.0f;                  // self-loop fill_value = 1.0
}

// deg -> dis in place; agg1 = dis^2 * x (conv1 self-loop message)
__global__ void k_dis_agg1(const float* __restrict__ x,
                           float* __restrict__ deg_dis,
                           float* __restrict__ agg1, int n) {
    int i = blockIdx.x * blockDim.x + threadIdx.x;
    if (i >= n) return;
    float d = deg_dis[i];
    float y = 0.0f;
    if (d > 0.0f) {
        y = rsqrtf(d);
        y = y * fmaf(-0.5f * d * y, y, 1.5f);  // Newton step -> full fp32
    }
    deg_dis[i] = y;
    agg1[i] = y * y * x[i];
}

// z = sum_k relu(agg1*W1[k]+b1[k])*W2[k];  agg2 = dis^2 * z (conv2 self-loop)
__global__ void k_mlp(const float* __restrict__ agg1,
                      const float* __restrict__ dis,
                      const float* __restrict__ W1, const float* __restrict__ b1,
                      const float* __restrict__ W2,
                      float* __restrict__ z, float* __restrict__ agg2, int n) {
    int i = blockIdx.x * blockDim.x + threadIdx.x;
    if (i >= n) return;
    float a = agg1[i];
    float acc = 0.0f;
#pragma unroll
    for (int k = 0; k < 4; ++k) {
        float h = fmaf(a, W1[k], b1[k]);
        h = fmaxf(h, 0.0f);
        acc = fmaf(h, W2[k], acc);
    }
    float d = dis[i];
    z[i]    = acc;
    agg2[i] = d * d * acc;
}

__global__ void k_final(const float* __restrict__ agg2,
                        const float* __restrict__ b2, const float* __restrict__ Wl,
                        const float* __restrict__ bl,
                        float* __restrict__ out, int n) {
    int i = blockIdx.x * blockDim.x + threadIdx.x;
    if (i >= n) return;
    float v = fmaf(agg2[i] + b2[0], Wl[0], bl[0]);
    out[i] = 1.0f / (1.0f + expf(-v));
}

// ---------------- edge kernels (grid-stride, 4 edges/thread) ----------------

__global__ void k_deg_accum(const long long* __restrict__ colp,
                            const float* __restrict__ ewp,
                            float* __restrict__ deg, long long E) {
    const long long stride = (long long)gridDim.x * blockDim.x * 4;
    long long e = ((long long)blockIdx.x * blockDim.x + threadIdx.x) * 4;
    for (; e + 3 < E; e += stride) {
        v2i64 c01 = ldnt2(colp + e);           // global_load_b128 th:NT
        v2i64 c23 = ldnt2(colp + e + 2);
        v4f   w   = ldnt4f(ewp + e);
        __builtin_prefetch(colp + e + stride, 0, 1);   // global_prefetch_b8
        __builtin_prefetch(ewp  + e + stride, 0, 1);
        atomicAdd(deg + (int)c01.x, w.x);      // global_atomic_add_f32 (L2)
        atomicAdd(deg + (int)c01.y, w.y);
        atomicAdd(deg + (int)c23.x, w.z);
        atomicAdd(deg + (int)c23.y, w.w);
    }
    for (; e < E; ++e) atomicAdd(deg + (int)colp[e], ewp[e]);
}

// dst[col] += dis[row]*ew*dis[col]*src[row]
// Edge weights staged via CDNA5 async-to-LDS (b128, ASYNCcnt); the NT b128
// index loads issued in between overlap the async copy's latency.
__global__ void k_edge_agg(const long long* __restrict__ rowp,
                           const long long* __restrict__ colp,
                           const float* __restrict__ ewp,
                           const float* __restrict__ dis,
                           const float* __restrict__ src,
                           float* __restrict__ dst, long long E) {
    __shared__ v4f sew[BLK];
    const unsigned ldsa = (unsigned)(size_t)(&sew[threadIdx.x]);  // LDS byte addr
    const long long stride = (long long)gridDim.x * blockDim.x * 4;
    long long e = ((long long)blockIdx.x * blockDim.x + threadIdx.x) * 4;
    for (; e + 3 < E; e += stride) {
        asm volatile("global_load_async_to_lds_b128 %0, %1, off"
                     :: "v"(ldsa), "v"((unsigned long long)(ewp + e)) : "memory");
        v2i64 r01 = ldnt2(rowp + e);
        v2i64 r23 = ldnt2(rowp + e + 2);
        v2i64 c01 = ldnt2(colp + e);
        v2i64 c23 = ldnt2(colp + e + 2);
        __builtin_prefetch(rowp + e + stride, 0, 1);
        __builtin_prefetch(colp + e + stride, 0, 1);
        int r0 = (int)r01.x, r1 = (int)r01.y, r2 = (int)r23.x, r3 = (int)r23.y;
        int c0 = (int)c01.x, c1 = (int)c01.y, c2 = (int)c23.x, c3 = (int)c23.y;
        // L2-resident gathers (dis/src are 4 MB arrays)
        float p0 = dis[r0] * dis[c0] * src[r0];
        float p1 = dis[r1] * dis[c1] * src[r1];
        float p2 = dis[r2] * dis[c2] * src[r2];
        float p3 = dis[r3] * dis[c3] * src[r3];
        asm volatile("s_wait_asynccnt 0" ::: "memory");
        v4f w = sew[threadIdx.x];              // ds_load_b128
        atomicAdd(dst + c0, p0 * w.x);
        atomicAdd(dst + c1, p1 * w.y);
        atomicAdd(dst + c2, p2 * w.z);
        atomicAdd(dst + c3, p3 * w.w);
    }
    for (; e < E; ++e) {                       // E%4 tail (E=16M -> empty)
        int r = (int)rowp[e], c = (int)colp[e];
        atomicAdd(dst + c, dis[r] * ewp[e] * dis[c] * src[r]);
    }
}

// ---------------- launch ----------------

extern "C" void kernel_launch(void* const* d_in, const int* in_sizes, int n_in,
                              void* d_out, int out_size, void* d_ws, size_t ws_size,
                              hipStream_t stream) {
    const float*     x  = (const float*)d_in[0];
    const long long* ei = (const long long*)d_in[1];   // int64 [2, E]
    const float*     ew = (const float*)d_in[2];
    const float*     W1 = (const float*)d_in[3];
    const float*     b1 = (const float*)d_in[4];
    const float*     W2 = (const float*)d_in[5];
    const float*     b2 = (const float*)d_in[6];
    const float*     Wl = (const float*)d_in[7];
    const float*     bl = (const float*)d_in[8];
    float* out = (float*)d_out;

    const int       N = in_sizes[0];
    const long long E = (long long)in_sizes[2];
    const long long* rowp = ei;
    const long long* colp = ei + E;

    float* deg_dis = (float*)d_ws;   // N floats: deg, then dis (in place)
    float* agg1    = deg_dis + N;    // N floats
    float* z       = agg1 + N;       // N floats
    float* agg2    = z + N;          // N floats  (16 MB total workspace)

    int nodeBlocks = (N + BLK - 1) / BLK;
    long long quads = (E + 3) / 4;
    long long eb = (quads + BLK - 1) / BLK;
    int edgeBlocks = (int)(eb < 2048 ? eb : 2048);

    k_init_deg <<<nodeBlocks, BLK, 0, stream>>>(deg_dis, N);
    k_deg_accum<<<edgeBlocks, BLK, 0, stream>>>(colp, ew, deg_dis, E);
    k_dis_agg1 <<<nodeBlocks, BLK, 0, stream>>>(x, deg_dis, agg1, N);
    k_edge_agg <<<edgeBlocks, BLK, 0, stream>>>(rowp, colp, ew, deg_dis, x, agg1, E);
    k_mlp      <<<nodeBlocks, BLK, 0, stream>>>(agg1, deg_dis, W1, b1, W2, z, agg2, N);
    k_edge_agg <<<edgeBlocks, BLK, 0, stream>>>(rowp, colp, ew, deg_dis, z, agg2, E);
    k_final    <<<nodeBlocks, BLK, 0, stream>>>(agg2, b2, Wl, bl, out, N);
}